// RRWHM_51118700757114
// MI455X (gfx1250) — compile-verified
//
#include <hip/hip_runtime.h>
#include <hip/hip_bf16.h>
#include <cstddef>

typedef __attribute__((ext_vector_type(2))) float v2f;
typedef __attribute__((ext_vector_type(8))) float v8f;

#define MN 144
#define NS 12
#define FLATK (MN * MN)   // 20736
#define B_ 16
#define ZSPLIT 3          // j-dimension split across gridDim.z
#define SZ (B_ * MN)      // one pw slice
#define MAX_ITER 50
#define SK_ITER 20
#define ALPHA 0.2f
#define BETA 30.0f
#define SK_EPS 1e-4f
#define TOLSQ 1e-10f      // TOL^2, compare sum of squares

// ---------------------------------------------------------------- init
__global__ void rrwhm_init(const int* __restrict__ ns_src,
                           const int* __restrict__ ns_tgt,
                           float* __restrict__ v,
                           float* __restrict__ diffb,
                           int* __restrict__ done) {
  const int tid = threadIdx.x;
  if (tid == 0) *done = 0;
  if (tid < B_) diffb[tid] = 0.f;
  for (int idx = tid; idx < B_ * MN; idx += blockDim.x) {
    const int b = idx / MN, r = idx % MN;
    const int t = r / NS, s = r % NS;           // v0 laid out [tgt, src] flat
    const int ns = ns_src[b], nt = ns_tgt[b];
    v[idx] = (s < ns && t < nt) ? 1.f / (float)(ns * nt) : 0.f;
  }
}

// ------------------------------------------------------- degree pass 1
__global__ __launch_bounds__(256) void rrwhm_degree(const float* __restrict__ H,
                                                    float* __restrict__ dtmp) {
  const int i = blockIdx.x, b = blockIdx.y, tid = threadIdx.x;
  const float* __restrict__ p = H + ((size_t)(b * MN + i)) * FLATK;
  float s = 0.f;
  for (int t = tid; t < FLATK; t += 256) s += p[t];   // 81 per thread
  __shared__ float r[256];
  r[tid] = s;
  __syncthreads();
  for (int off = 128; off > 0; off >>= 1) {
    if (tid < off) r[tid] += r[tid + off];
    __syncthreads();
  }
  if (tid == 0) dtmp[b * MN + i] = r[0];
}

// ------------------------------------------------------- degree pass 2
__global__ __launch_bounds__(256) void rrwhm_maxdeg(const float* __restrict__ dtmp,
                                                    float* __restrict__ invD) {
  const int b = blockIdx.x, tid = threadIdx.x;
  __shared__ float r[MN];
  if (tid < MN) r[tid] = dtmp[b * MN + tid];
  __syncthreads();
  if (tid == 0) {
    float mx = r[0];
    for (int i = 1; i < MN; ++i) mx = fmaxf(mx, r[i]);
    invD[b] = 1.f / mx;
  }
}

// ----------------------------------------------- tensor power iteration
// pwpart[z][b, i0..i0+15] = invD[b] * sum_{j in z-slice, k} H[b,i,j,k] v[j] v[k]
// V_WMMA_F32_16X16X4_F32, B operand = v[j]*v[k..] replicated over all 16 cols.
// Grid (9, 16, 3), 512 threads = 16 waves; wave owns 3 contiguous j-rows.
__global__ __launch_bounds__(512) void rrwhm_power(const float* __restrict__ H,
                                                   const float* __restrict__ v,
                                                   const float* __restrict__ invD,
                                                   float* __restrict__ pwpart,
                                                   const int* __restrict__ done) {
  if (*done) return;
  const int b = blockIdx.y;
  const int i0 = blockIdx.x * 16;
  const int z = blockIdx.z;           // 0..2
  const int tid = threadIdx.x;
  const int lane = tid & 31;
  const int wave = tid >> 5;          // 0..15
  const int half = (lane >> 4) & 1;   // K sub-pair selector
  const int m = lane & 15;            // A-matrix row within tile

  __shared__ float sv[MN];
  __shared__ float red[16][16];

  if (tid < MN) sv[tid] = v[b * MN + tid];
  __syncthreads();

  const int jbase = z * (MN / ZSPLIT) + wave * 3;   // 3 j-rows per wave
  const float* __restrict__ Hrow =
      H + ((size_t)(b * MN + i0 + m)) * FLATK + 2 * half;
  const float* __restrict__ svh = sv + 2 * half;

  v8f acc0 = {}, acc1 = {}, acc2 = {}, acc3 = {};
  for (int jj = 0; jj < 3; ++jj) {
    const int j = jbase + jj;
    const float vj = sv[j];
    const float* __restrict__ hp = Hrow + j * MN;
    // 9 quads of 16 k-values -> 4 WMMAs each, 4 independent accumulators.
    for (int q = 0; q < MN / 16; ++q) {
      const int kk = q * 16;
      __builtin_prefetch(hp + kk + 1024, 0, 0);   // global_prefetch_b8, 4KB ahead
      v2f a0 = *(const v2f*)(hp + kk);
      v2f a1 = *(const v2f*)(hp + kk + 4);
      v2f a2 = *(const v2f*)(hp + kk + 8);
      v2f a3 = *(const v2f*)(hp + kk + 12);
      v2f b0; b0[0] = vj * svh[kk];      b0[1] = vj * svh[kk + 1];
      v2f b1; b1[0] = vj * svh[kk + 4];  b1[1] = vj * svh[kk + 5];
      v2f b2; b2[0] = vj * svh[kk + 8];  b2[1] = vj * svh[kk + 9];
      v2f b3; b3[0] = vj * svh[kk + 12]; b3[1] = vj * svh[kk + 13];
      acc0 = __builtin_amdgcn_wmma_f32_16x16x4_f32(false, a0, false, b0,
                                                   (short)0, acc0, false, false);
      acc1 = __builtin_amdgcn_wmma_f32_16x16x4_f32(false, a1, false, b1,
                                                   (short)0, acc1, false, false);
      acc2 = __builtin_amdgcn_wmma_f32_16x16x4_f32(false, a2, false, b2,
                                                   (short)0, acc2, false, false);
      acc3 = __builtin_amdgcn_wmma_f32_16x16x4_f32(false, a3, false, b3,
                                                   (short)0, acc3, false, false);
    }
  }
  v8f acc = (acc0 + acc1) + (acc2 + acc3);

  // D layout: lanes 0-15 -> N=lane, M=r; lanes 16-31 -> M=8+r. All columns
  // equal (B replicated), so read column N=0 (lanes 0 and 16).
  if ((lane & 15) == 0) {
#pragma unroll
    for (int r = 0; r < 8; ++r) red[wave][half * 8 + r] = acc[r];
  }
  __syncthreads();
  if (tid < 16) {
    float s = 0.f;
#pragma unroll
    for (int w = 0; w < 16; ++w) s += red[w][tid];   // deterministic order
    pwpart[(size_t)z * SZ + b * MN + i0 + tid] = s * invD[b];
  }
}

// ---------------------------------- normalize + sinkhorn + relax update
__global__ __launch_bounds__(256) void rrwhm_update(const float* __restrict__ pwpart,
                                                    float* __restrict__ v,
                                                    const int* __restrict__ ns_src,
                                                    const int* __restrict__ ns_tgt,
                                                    float* __restrict__ diffb,
                                                    const int* __restrict__ done) {
  if (*done) return;
  const int b = blockIdx.x, tid = threadIdx.x;

  __shared__ float sp[MN];           // pw (degree-normalized)
  __shared__ float vn[MN];
  __shared__ float smat[NS][NS];
  __shared__ float norm[NS];
  __shared__ float scal[4];
  __shared__ float acc[MN];

  if (tid < MN) {                    // combine the 3 z-partials deterministically
    const int idx = b * MN + tid;
    sp[tid] = pwpart[idx] + pwpart[SZ + idx] + pwpart[2 * SZ + idx];
  }
  __syncthreads();
  if (tid == 0) {
    float s = 0.f;
    for (int i = 0; i < MN; ++i) s += fabsf(sp[i]);
    scal[0] = s;
  }
  __syncthreads();
  if (tid < MN) vn[tid] = sp[tid] / scal[0];
  __syncthreads();
  if (tid < MN) {                    // s[src][tgt] = vn[tgt*12+src]
    const int src = tid / NS, tgt = tid % NS;
    smat[src][tgt] = vn[tgt * NS + src];
  }
  __syncthreads();
  if (tid == 0) {
    const float* f = &smat[0][0];
    float mx = f[0];
    for (int i = 1; i < MN; ++i) mx = fmaxf(mx, f[i]);
    scal[1] = mx;
  }
  __syncthreads();
  if (tid < MN) {
    const int src = tid / NS, tgt = tid % NS;
    smat[src][tgt] = expf(BETA * smat[src][tgt] / scal[1]) + SK_EPS;
  }
  __syncthreads();

  const int ns = ns_src[b], nt = ns_tgt[b];
  for (int it = 0; it < SK_ITER; ++it) {
    if ((it & 1) == 0) {             // per-column normalizer over valid rows
      if (tid < NS) {
        float c = 0.f;
        for (int s = 0; s < NS; ++s)
          if (s < ns) c += smat[s][tid];
        norm[tid] = (c > 0.f) ? c : 1.f;
      }
      __syncthreads();
      if (tid < MN) {
        const int src = tid / NS, tgt = tid % NS;
        const bool ok = (src < ns) && (tgt < nt);
        smat[src][tgt] = ok ? smat[src][tgt] / norm[tgt] : 0.f;
      }
    } else {                         // per-row normalizer over valid cols
      if (tid < NS) {
        float c = 0.f;
        for (int t = 0; t < NS; ++t)
          if (t < nt) c += smat[tid][t];
        norm[tid] = (c > 0.f) ? c : 1.f;
      }
      __syncthreads();
      if (tid < MN) {
        const int src = tid / NS, tgt = tid % NS;
        const bool ok = (src < ns) && (tgt < nt);
        smat[src][tgt] = ok ? smat[src][tgt] / norm[src] : 0.f;
      }
    }
    __syncthreads();
  }

  if (tid < MN) {                    // v2 = a*T(sk).flat + (1-a)*vn
    const int s2 = tid % NS, t2 = tid / NS;
    acc[tid] = ALPHA * smat[s2][t2] + (1.f - ALPHA) * vn[tid];
  }
  __syncthreads();
  if (tid == 0) {
    float s = 0.f;
    for (int i = 0; i < MN; ++i) s += fabsf(acc[i]);
    scal[2] = s;
  }
  __syncthreads();
  if (tid < MN) {
    const float v2 = acc[tid] / scal[2];
    v[b * MN + tid] = v2;
    const float dd = v2 - sp[tid];
    acc[tid] = dd * dd;
  }
  __syncthreads();
  if (tid == 0) {
    float s = 0.f;
    for (int i = 0; i < MN; ++i) s += acc[i];
    diffb[b] = s;                    // deterministic per-batch partial
  }
}

// ------------------------------------------------- global convergence
__global__ void rrwhm_check(const float* __restrict__ diffb,
                            int* __restrict__ done) {
  float s = 0.f;
  for (int i = 0; i < B_; ++i) s += diffb[i];
  if (s < TOLSQ) *done = 1;          // once set, stays set (stale diffb ok)
}

// ------------------------------------------------------------ copy out
__global__ void rrwhm_copy(const float* __restrict__ v, float* __restrict__ out) {
  const int idx = blockIdx.x * 256 + threadIdx.x;
  if (idx < B_ * MN) out[idx] = v[idx];
}

extern "C" void kernel_launch(void* const* d_in, const int* in_sizes, int n_in,
                              void* d_out, int out_size, void* d_ws, size_t ws_size,
                              hipStream_t stream) {
  const float* H = (const float*)d_in[0];
  // d_in[1] = num_src (always 12, hardcoded)
  const int* ns_src = (const int*)d_in[2];
  const int* ns_tgt = (const int*)d_in[3];

  float* ws = (float*)d_ws;
  float* dtmp   = ws;                    // 2304
  float* invD   = ws + 2304;             // 16
  float* v      = ws + 2320;             // 2304
  float* pwpart = ws + 4624;             // 3 * 2304
  float* diffb  = ws + 4624 + 3 * SZ;    // 16
  int*   done   = (int*)(ws + 4640 + 3 * SZ);

  rrwhm_init<<<1, 256, 0, stream>>>(ns_src, ns_tgt, v, diffb, done);
  rrwhm_degree<<<dim3(MN, B_), 256, 0, stream>>>(H, dtmp);
  rrwhm_maxdeg<<<B_, 256, 0, stream>>>(dtmp, invD);

  for (int iter = 0; iter < MAX_ITER; ++iter) {
    rrwhm_power<<<dim3(MN / 16, B_, ZSPLIT), 512, 0, stream>>>(H, v, invD, pwpart, done);
    rrwhm_update<<<B_, 256, 0, stream>>>(pwpart, v, ns_src, ns_tgt, diffb, done);
    rrwhm_check<<<1, 1, 0, stream>>>(diffb, done);
  }
  rrwhm_copy<<<(B_ * MN + 255) / 256, 256, 0, stream>>>(v, (float*)d_out);
}